// CensusConsistencyLayer_26147760898487
// MI455X (gfx1250) — compile-verified
//
#include <hip/hip_runtime.h>

// CensusConsistencyLayer for MI455X (gfx1250, wave32).
// Shapes fixed by the reference harness: B=16, C=1, H=W=1024, M=1024.
//
// Bandwidth-bound (AI ~ 0.1 FLOP/byte): target the 23.3 TB/s HBM roofline.
// p + ids = 134 MB fit in the 192 MB L2, so pass 1 warms L2 and pass 3
// re-reads from it; pass-3 loads/stores use non-temporal hints so the 67 MB
// output stream does not evict anything useful. Per-WGP LDS histograms
// (ds_add_f32) compress 16.7M scatter-adds into ~2M global float atomics on
// a 64 KB L2-resident accumulator.

typedef float fvec4 __attribute__((ext_vector_type(4)));
typedef int   ivec4 __attribute__((ext_vector_type(4)));

namespace {
constexpr int   kB   = 16;
constexpr int   kHW  = 1024 * 1024;
constexpr int   kM   = 1024;
constexpr float kEps = 1e-6f;

constexpr int kThreads        = 256;                     // 8 wave32 waves
constexpr int kBlocksPerBatch = 128;
constexpr int kChunk          = kHW / kBlocksPerBatch;   // 8192 pixels / block
}  // namespace

// ---------------------------------------------------------------------------
// Pass 0: zero the per-(batch, admin) accumulator (d_ws is poisoned 0xAA).
// ---------------------------------------------------------------------------
__global__ void cc_zero_S(float* __restrict__ S) {
    int i = blockIdx.x * blockDim.x + threadIdx.x;
    if (i < kB * kM) S[i] = 0.0f;
}

// ---------------------------------------------------------------------------
// Pass 1: segment sum.  Each block owns an 8192-pixel stripe of one batch.
// LDS histogram (1024 f32 = 4 KB of the 320 KB WGP LDS) via ds_add_f32,
// then a sparse flush with global_atomic_add_f32.  Default (RT) loads warm
// the L2 for pass 3.
// ---------------------------------------------------------------------------
__global__ __launch_bounds__(kThreads)
void cc_seg_sum(const float* __restrict__ p,
                const int*   __restrict__ ids,
                float*       __restrict__ S) {
    __shared__ float hist[kM];
    for (int i = threadIdx.x; i < kM; i += kThreads) hist[i] = 0.0f;
    __syncthreads();

    const int    b    = blockIdx.y;
    const size_t base = (size_t)b * kHW + (size_t)blockIdx.x * kChunk;
    const fvec4* __restrict__ p4  = reinterpret_cast<const fvec4*>(p + base);
    const ivec4* __restrict__ id4 = reinterpret_cast<const ivec4*>(ids + base);

    constexpr int kNVec = kChunk / 4;   // 2048 float4s per block
    // Two b128 loads of each stream in flight per iteration.
    for (int i = threadIdx.x; i < kNVec; i += 2 * kThreads) {
        const int   j   = i + kThreads;            // always < kNVec (2048=8*256)
        const fvec4 pv0 = p4[i];                   // global_load_b128
        const fvec4 pv1 = p4[j];
        const ivec4 iv0 = id4[i];
        const ivec4 iv1 = id4[j];
        if (iv0.x >= 0) atomicAdd(&hist[iv0.x], pv0.x);   // ds_add_f32
        if (iv0.y >= 0) atomicAdd(&hist[iv0.y], pv0.y);
        if (iv0.z >= 0) atomicAdd(&hist[iv0.z], pv0.z);
        if (iv0.w >= 0) atomicAdd(&hist[iv0.w], pv0.w);
        if (iv1.x >= 0) atomicAdd(&hist[iv1.x], pv1.x);
        if (iv1.y >= 0) atomicAdd(&hist[iv1.y], pv1.y);
        if (iv1.z >= 0) atomicAdd(&hist[iv1.z], pv1.z);
        if (iv1.w >= 0) atomicAdd(&hist[iv1.w], pv1.w);
    }
    __syncthreads();

    float* __restrict__ Sb = S + (size_t)b * kM;
    for (int i = threadIdx.x; i < kM; i += kThreads) {
        const float v = hist[i];
        if (v != 0.0f) atomicAdd(&Sb[i], v);       // global_atomic_add_f32
    }
}

// ---------------------------------------------------------------------------
// Pass 2: scale table  scale[b,m] = census[b,m] / (S[b,m] + eps).  16K elems.
// ---------------------------------------------------------------------------
__global__ void cc_make_scale(const float* __restrict__ S,
                              const float* __restrict__ census,
                              float*       __restrict__ scale) {
    int i = blockIdx.x * blockDim.x + threadIdx.x;
    if (i < kB * kM) scale[i] = census[i] / (S[i] + kEps);
}

// ---------------------------------------------------------------------------
// Pass 3: out = valid ? p * scale[b, id] : 0.  Streaming pass: non-temporal
// loads (last use of p/ids; still hit L2, no re-retention) and non-temporal
// stores (write-once 67 MB output must not thrash L2).  Scale table (64 KB)
// gathered with regular loads -> stays L2/WGP$-resident.
// ---------------------------------------------------------------------------
__global__ __launch_bounds__(kThreads)
void cc_apply(const float* __restrict__ p,
              const int*   __restrict__ ids,
              const float* __restrict__ scale,
              float*       __restrict__ out) {
    const size_t v  = (size_t)blockIdx.x * kThreads + threadIdx.x; // vec4 idx
    const size_t px = v * 4;                    // kHW % 4 == 0 -> one batch
    const int    b  = (int)(px >> 20);          // px / kHW (kHW = 2^20)

    const fvec4 pv = __builtin_nontemporal_load(reinterpret_cast<const fvec4*>(p) + v);
    const ivec4 iv = __builtin_nontemporal_load(reinterpret_cast<const ivec4*>(ids) + v);
    const float* __restrict__ sb = scale + (size_t)b * kM;

    fvec4 ov;
    ov.x = (iv.x >= 0) ? pv.x * sb[iv.x] : 0.0f;
    ov.y = (iv.y >= 0) ? pv.y * sb[iv.y] : 0.0f;
    ov.z = (iv.z >= 0) ? pv.z * sb[iv.z] : 0.0f;
    ov.w = (iv.w >= 0) ? pv.w * sb[iv.w] : 0.0f;
    __builtin_nontemporal_store(ov, reinterpret_cast<fvec4*>(out) + v);
}

// ---------------------------------------------------------------------------
extern "C" void kernel_launch(void* const* d_in, const int* in_sizes, int n_in,
                              void* d_out, int out_size, void* d_ws, size_t ws_size,
                              hipStream_t stream) {
    (void)in_sizes; (void)n_in; (void)out_size; (void)ws_size;

    const float* P      = reinterpret_cast<const float*>(d_in[0]); // (B,1,H,W)
    const int*   ids    = reinterpret_cast<const int*>(d_in[1]);   // (B,H,W)
    const float* census = reinterpret_cast<const float*>(d_in[2]); // (B,M)
    float*       out    = reinterpret_cast<float*>(d_out);

    float* S     = reinterpret_cast<float*>(d_ws);   // B*M f32 = 64 KB
    float* scale = S + (size_t)kB * kM;              // B*M f32 = 64 KB

    // Pass 0: zero accumulator.
    cc_zero_S<<<(kB * kM + 255) / 256, 256, 0, stream>>>(S);

    // Pass 1: LDS-staged segment sum.
    dim3 grid1(kBlocksPerBatch, kB);
    cc_seg_sum<<<grid1, kThreads, 0, stream>>>(P, ids, S);

    // Pass 2: scale table.
    cc_make_scale<<<(kB * kM + 255) / 256, 256, 0, stream>>>(S, census, scale);

    // Pass 3: gather + multiply, streaming (NT).
    const size_t nvec   = (size_t)kB * kHW / 4;      // 4,194,304 vec4s
    const int    blocks = (int)((nvec + kThreads - 1) / kThreads);
    cc_apply<<<blocks, kThreads, 0, stream>>>(P, ids, scale, out);
}